// ConsistencyLoss_70068096467696
// MI455X (gfx1250) — compile-verified
//
#include <hip/hip_runtime.h>
#include <math.h>

typedef __attribute__((ext_vector_type(2))) float v2f;
typedef __attribute__((ext_vector_type(8))) float v8f;

#define L        128
#define D        262144
#define NSPLIT   128
#define KSPLIT   (D / NSPLIT)      // 2048 K per workgroup
#define KC       64                // K-chunk staged in LDS per iteration
#define NCHUNK   (KSPLIT / KC)     // 32
#define EPSV     1e-6f

// ---------------------------------------------------------------------------
// Stage 1: split-K Gram partials via V_WMMA_F32_16X16X4_F32.
// LDS holds a 128 x KC fp32 slice, pair-interleaved so that the WMMA A/B
// fragment (K=2*half, 2*half+1 per lane) is one ds_load_b64:
//   lds2[p*L + row] = { S[row][2p], S[row][2p+1] }  (k local to chunk)
// ---------------------------------------------------------------------------
__global__ __launch_bounds__(256) void gram_stage1(const float* __restrict__ slots,
                                                   float* __restrict__ part) {
  __shared__ float lds[KC * L];                       // 32 KB
  float2* lds2 = (float2*)lds;

  const int tid   = threadIdx.x;
  const int wave  = tid >> 5;                         // 8 waves (wave32)
  const int lane  = tid & 31;
  const int hl    = lane >> 4;                        // lane half (0/1)
  const int l16   = lane & 15;
  const int split = blockIdx.x;
  const long kbase = (long)split * KSPLIT;

  v8f acc[8];
#pragma unroll
  for (int c = 0; c < 8; ++c) acc[c] = (v8f)0.f;

  // Each thread stages 8 float4s per chunk: idx -> (row, kq) with 16 float4s/row.
  float4 buf[8];
  int rowv[8], kqv[8];
#pragma unroll
  for (int j = 0; j < 8; ++j) {
    int idx = tid + 256 * j;                          // 0..2047
    rowv[j] = idx >> 4;
    kqv[j]  = idx & 15;
    buf[j]  = *(const float4*)(slots + (long)rowv[j] * D + kbase + kqv[j] * 4);
  }

  for (int chunk = 0; chunk < NCHUNK; ++chunk) {
    __syncthreads();                                  // previous compute done
#pragma unroll
    for (int j = 0; j < 8; ++j) {
      const int r = rowv[j], kq = kqv[j];
      lds2[(kq * 2 + 0) * L + r] = make_float2(buf[j].x, buf[j].y);
      lds2[(kq * 2 + 1) * L + r] = make_float2(buf[j].z, buf[j].w);
    }
    __syncthreads();

    // Software pipeline: issue next chunk's global loads, then compute from LDS.
    if (chunk + 1 < NCHUNK) {
      const long k0n = kbase + (long)(chunk + 1) * KC;
#pragma unroll
      for (int j = 0; j < 8; ++j)
        buf[j] = *(const float4*)(slots + (long)rowv[j] * D + k0n + kqv[j] * 4);
    }

#pragma unroll
    for (int kk = 0; kk < KC; kk += 4) {
      const int p = (kk >> 1) + hl;                   // pair index for this lane half
      const float2 af = lds2[p * L + wave * 16 + l16];
      v2f a; a.x = af.x; a.y = af.y;                  // A: rows of this wave's strip
#pragma unroll
      for (int c = 0; c < 8; ++c) {
        const float2 bf = lds2[p * L + c * 16 + l16]; // B: same layout, column tile c
        v2f b; b.x = bf.x; b.y = bf.y;
        acc[c] = __builtin_amdgcn_wmma_f32_16x16x4_f32(
            /*neg_a=*/false, a, /*neg_b=*/false, b,
            /*c_mod=*/(short)0, acc[c], /*reuse_a=*/false, /*reuse_b=*/false);
      }
    }
  }

  // C/D layout: VGPR r, lane -> M = r + 8*hl, N = l16 (per tile).
  float* out = part + (long)split * (L * L);
#pragma unroll
  for (int c = 0; c < 8; ++c) {
#pragma unroll
    for (int r = 0; r < 8; ++r) {
      const int row = wave * 16 + hl * 8 + r;
      const int col = c * 16 + l16;
      out[row * L + col] = acc[c][r];
    }
  }
}

// ---------------------------------------------------------------------------
// Stage 2: deterministic fixed-order reduction of the NSPLIT partials.
// ---------------------------------------------------------------------------
__global__ __launch_bounds__(256) void gram_reduce(const float* __restrict__ part,
                                                   float* __restrict__ gram) {
  const int e = blockIdx.x * 256 + threadIdx.x;       // 0..16383
  float s = 0.f;
  for (int sp = 0; sp < NSPLIT; ++sp)
    s += part[(long)sp * (L * L) + e];
  gram[e] = s;
}

// ---------------------------------------------------------------------------
// Stage 3: loss from the 128x128 Gram matrix (one block, thread per row).
// ---------------------------------------------------------------------------
__global__ __launch_bounds__(128) void loss_kernel(const float* __restrict__ gram,
                                                   const float* __restrict__ temperature,
                                                   float* __restrict__ out) {
  __shared__ float norms[L];
  __shared__ float rowacc[L];
  const int i = threadIdx.x;
  const float invT = 1.f / temperature[0];

  norms[i] = sqrtf(gram[i * L + i]);
  __syncthreads();
  const float ni = norms[i];

  float sumE = 0.f;
  for (int j = 0; j < L; ++j) {
    const float sim = gram[i * L + j] / fmaxf(ni * norms[j], EPSV);
    sumE += expf(sim * invT);
  }
  float a = 0.f;
  for (int j = i + 1; j < L; ++j) {
    const float sim = gram[i * L + j] / fmaxf(ni * norms[j], EPSV);
    const float lg  = sim * invT;
    const float E   = expf(lg);
    a += -(lg - logf(sumE - E)) * (float)(j - i);
  }
  rowacc[i] = a;
  __syncthreads();
  if (i == 0) {
    float t = 0.f;
    for (int k = 0; k < L; ++k) t += rowacc[k];
    out[0] = t / (((float)(L - 1) * (float)(L - 1)) * 0.5f);
  }
}

// ---------------------------------------------------------------------------
extern "C" void kernel_launch(void* const* d_in, const int* in_sizes, int n_in,
                              void* d_out, int out_size, void* d_ws, size_t ws_size,
                              hipStream_t stream) {
  (void)in_sizes; (void)n_in; (void)out_size; (void)ws_size;
  const float* slots       = (const float*)d_in[0];
  const float* temperature = (const float*)d_in[1];

  float* part = (float*)d_ws;                         // NSPLIT * 128*128 floats (8 MB)
  float* gram = part + (size_t)NSPLIT * L * L;        // + 128*128 floats (64 KB)

  gram_stage1<<<NSPLIT, 256, 0, stream>>>(slots, part);
  gram_reduce<<<(L * L) / 256, 256, 0, stream>>>(part, gram);
  loss_kernel<<<1, L, 0, stream>>>(gram, temperature, (float*)d_out);
}